// AutoRegMLAttention_936302871131
// MI455X (gfx1250) — compile-verified
//
#include <hip/hip_runtime.h>
#include <math.h>

typedef float v2f __attribute__((ext_vector_type(2)));
typedef float v8f __attribute__((ext_vector_type(8)));

#define NUM_HEADS 32
#define HEAD_DIM  128
#define ROPE_DIM  64
#define HIDDEN    4096
#define C_KV      512
#define C_Q       1536
#define CACHE_LEN 4096
#define BATCH     4
#define S_NEW     (CACHE_LEN + 1)          // 4097 positions
#define S_PAD     4112                     // 4097 rounded up to 16
#define Q_DIM     (C_KV + ROPE_DIM)        // 576 absorbed-q dim

__device__ __forceinline__ v8f wmma_f32(v2f a, v2f b, v8f c) {
    return __builtin_amdgcn_wmma_f32_16x16x4_f32(false, a, false, b, (short)0, c, false, false);
}

static constexpr float SCALE = 0.07216878364870323f;   // 1/sqrt(192)

// ---------------------------------------------------------------------------
// Kernel 1: down projections. c_KV (->new cache row), c_Q (->ws), k_R + RoPE
// (->new cache row). Grid 9 blocks x 256: blk 0-1 = W_DKV cols, 2-7 = W_DQ,
// 8 = W_KR + rope. Hidden states (64KB) staged in LDS, each weight read once.
// ---------------------------------------------------------------------------
__global__ void k_down(const float* __restrict__ hs,
                       const float* __restrict__ W_DKV,
                       const float* __restrict__ W_DQ,
                       const float* __restrict__ W_KR,
                       float* __restrict__ cq_ws,
                       float* __restrict__ out_ckv,   // [B][4097][512]
                       float* __restrict__ out_kr) {  // [B][4097][64]
    __shared__ float h_lds[BATCH * HIDDEN];
    __shared__ float kr_raw[BATCH * ROPE_DIM];
    for (int i = threadIdx.x; i < BATCH * HIDDEN; i += 256) h_lds[i] = hs[i];
    __syncthreads();

    const int blk = blockIdx.x;
    if (blk < 2) {                                   // c_KV: 512 cols
        const int col = blk * 256 + threadIdx.x;
        float acc[4] = {0.f, 0.f, 0.f, 0.f};
        for (int r = 0; r < HIDDEN; ++r) {
            const float w = W_DKV[(size_t)r * C_KV + col];
            #pragma unroll
            for (int b = 0; b < 4; ++b) acc[b] += h_lds[b * HIDDEN + r] * w;
        }
        #pragma unroll
        for (int b = 0; b < 4; ++b)
            out_ckv[((size_t)b * S_NEW + CACHE_LEN) * C_KV + col] = acc[b];
    } else if (blk < 8) {                            // c_Q: 1536 cols
        const int col = (blk - 2) * 256 + threadIdx.x;
        float acc[4] = {0.f, 0.f, 0.f, 0.f};
        for (int r = 0; r < HIDDEN; ++r) {
            const float w = W_DQ[(size_t)r * C_Q + col];
            #pragma unroll
            for (int b = 0; b < 4; ++b) acc[b] += h_lds[b * HIDDEN + r] * w;
        }
        #pragma unroll
        for (int b = 0; b < 4; ++b) cq_ws[(size_t)b * C_Q + col] = acc[b];
    } else {                                         // k_R: 64 cols + rope
        if (threadIdx.x < ROPE_DIM) {
            const int col = threadIdx.x;
            float acc[4] = {0.f, 0.f, 0.f, 0.f};
            for (int r = 0; r < HIDDEN; ++r) {
                const float w = W_KR[(size_t)r * ROPE_DIM + col];
                #pragma unroll
                for (int b = 0; b < 4; ++b) acc[b] += h_lds[b * HIDDEN + r] * w;
            }
            #pragma unroll
            for (int b = 0; b < 4; ++b) kr_raw[b * ROPE_DIM + col] = acc[b];
        }
        __syncthreads();
        // rope at position CACHE_LEN; 4 batches x 64 dims = 256 threads
        const int b = threadIdx.x >> 6, d = threadIdx.x & 63, i = d >> 1;
        const float ang = (float)CACHE_LEN * powf(10000.0f, -(2.0f * i) / (float)ROPE_DIM);
        const float cs = cosf(ang), sn = sinf(ang);
        const float x1 = kr_raw[b * ROPE_DIM + 2 * i];
        const float x2 = kr_raw[b * ROPE_DIM + 2 * i + 1];
        const float v = (d & 1) ? (x1 * sn + x2 * cs) : (x1 * cs - x2 * sn);
        out_kr[((size_t)b * S_NEW + CACHE_LEN) * ROPE_DIM + d] = v;
    }
}

// ---------------------------------------------------------------------------
// Cache copies (old rows -> new cache with stride 4097)
// ---------------------------------------------------------------------------
__global__ void k_copy_ckv(const float4* __restrict__ src, float4* __restrict__ dst) {
    const int idx = blockIdx.x * 256 + threadIdx.x;       // B*4096*128 items
    const int b = idx / (CACHE_LEN * 128);
    const int rem = idx - b * (CACHE_LEN * 128);
    const int pos = rem >> 7, q = rem & 127;
    dst[((size_t)b * S_NEW + pos) * 128 + q] = src[((size_t)b * CACHE_LEN + pos) * 128 + q];
}
__global__ void k_copy_kr(const float4* __restrict__ src, float4* __restrict__ dst) {
    const int idx = blockIdx.x * 256 + threadIdx.x;       // B*4096*16 items
    const int b = idx / (CACHE_LEN * 16);
    const int rem = idx - b * (CACHE_LEN * 16);
    const int pos = rem >> 4, q = rem & 15;
    dst[((size_t)b * S_NEW + pos) * 16 + q] = src[((size_t)b * CACHE_LEN + pos) * 16 + q];
}

// ---------------------------------------------------------------------------
// Kernel 2: per-head q pipeline. q_C = c_Q@W_UQ_h, q_R = rope(c_Q@W_QR_h),
// q_abs = q_C @ W_UK_h^T. Emits qcomb[b][h][576] = SCALE*[q_abs | q_R].
// Grid 32 (head) x 256.
// ---------------------------------------------------------------------------
__global__ void k_q(const float* __restrict__ cq_ws,
                    const float* __restrict__ W_UQ,
                    const float* __restrict__ W_QR,
                    const float* __restrict__ W_UK,
                    float* __restrict__ qcomb) {
    const int h = blockIdx.x;
    __shared__ float cq[BATCH * C_Q];        // 24KB
    __shared__ float qc[BATCH * HEAD_DIM];   // 2KB
    __shared__ float qr_raw[BATCH * ROPE_DIM];
    for (int i = threadIdx.x; i < BATCH * C_Q; i += 256) cq[i] = cq_ws[i];
    __syncthreads();

    {   // q_C: threads -> (d, batch pair)
        const int d = threadIdx.x & 127, b0 = threadIdx.x >> 7;
        float a0 = 0.f, a1 = 0.f;
        const float* wcol = W_UQ + (size_t)h * HEAD_DIM + d;
        for (int r = 0; r < C_Q; ++r) {
            const float w = wcol[(size_t)r * (NUM_HEADS * HEAD_DIM)];
            a0 += cq[b0 * C_Q + r] * w;
            a1 += cq[(b0 + 2) * C_Q + r] * w;
        }
        qc[b0 * HEAD_DIM + d] = a0;
        qc[(b0 + 2) * HEAD_DIM + d] = a1;
    }
    {   // q_R raw: 4 batches x 64 dims
        const int b = threadIdx.x >> 6, d = threadIdx.x & 63;
        float acc = 0.f;
        const float* wcol = W_QR + (size_t)h * ROPE_DIM + d;
        for (int r = 0; r < C_Q; ++r)
            acc += cq[b * C_Q + r] * wcol[(size_t)r * (NUM_HEADS * ROPE_DIM)];
        qr_raw[b * ROPE_DIM + d] = acc;
    }
    __syncthreads();
    {   // rope(q_R) -> qcomb tail, scaled
        const int b = threadIdx.x >> 6, d = threadIdx.x & 63, i = d >> 1;
        const float ang = (float)CACHE_LEN * powf(10000.0f, -(2.0f * i) / (float)ROPE_DIM);
        const float cs = cosf(ang), sn = sinf(ang);
        const float x1 = qr_raw[b * ROPE_DIM + 2 * i];
        const float x2 = qr_raw[b * ROPE_DIM + 2 * i + 1];
        const float v = (d & 1) ? (x1 * sn + x2 * cs) : (x1 * cs - x2 * sn);
        qcomb[((size_t)b * NUM_HEADS + h) * Q_DIM + C_KV + d] = v * SCALE;
    }
    // q_abs: thread handles c and c+256, 4 batches, W_UK row read once
    for (int cc = 0; cc < 2; ++cc) {
        const int c = threadIdx.x + cc * 256;
        const float* wrow = W_UK + (size_t)c * (NUM_HEADS * HEAD_DIM) + (size_t)h * HEAD_DIM;
        float acc[4] = {0.f, 0.f, 0.f, 0.f};
        for (int d = 0; d < HEAD_DIM; ++d) {
            const float w = wrow[d];
            #pragma unroll
            for (int b = 0; b < 4; ++b) acc[b] += qc[b * HEAD_DIM + d] * w;
        }
        #pragma unroll
        for (int b = 0; b < 4; ++b)
            qcomb[((size_t)b * NUM_HEADS + h) * Q_DIM + c] = acc[b] * SCALE;
    }
}

// ---------------------------------------------------------------------------
// Kernel 3: scores via WMMA f32 16x16x4. Per wave: one 16-position tile x all
// 32 heads (2 accumulators share the A fragment). A = latent-K rows, B = q.
// Grid (33, B) x 256 (8 waves).
// ---------------------------------------------------------------------------
__global__ void k_scores(const float* __restrict__ ckv,   // [B][4097][512]
                         const float* __restrict__ kr,    // [B][4097][64]
                         const float* __restrict__ qcomb, // [B][32][576]
                         float* __restrict__ sp) {        // [B][32][4112]
    const int b = blockIdx.y;
    const int wave = threadIdx.x >> 5, lane = threadIdx.x & 31;
    const int mtile = blockIdx.x * 8 + wave;
    if (mtile >= 257) return;                 // whole-wave uniform exit
    const int lhalf = lane & 15;
    const int koff = (lane >> 4) << 1;        // 0 or 2
    int row = mtile * 16 + lhalf;
    if (row > CACHE_LEN) row = CACHE_LEN;     // clamp: garbage never stored
    const float* ckv_row = ckv + ((size_t)b * S_NEW + row) * C_KV;
    const float* kr_row  = kr  + ((size_t)b * S_NEW + row) * ROPE_DIM;
    const float* q0 = qcomb + ((size_t)b * NUM_HEADS + lhalf) * Q_DIM;
    const float* q1 = qcomb + ((size_t)b * NUM_HEADS + 16 + lhalf) * Q_DIM;

    v8f c0 = {}; v8f c1 = {};
    for (int kb = 0; kb < C_KV; kb += 4) {
        const int k = kb + koff;
        v2f a;  a.x = ckv_row[k]; a.y = ckv_row[k + 1];
        v2f b0; b0.x = q0[k];     b0.y = q0[k + 1];
        v2f b1; b1.x = q1[k];     b1.y = q1[k + 1];
        c0 = wmma_f32(a, b0, c0);
        c1 = wmma_f32(a, b1, c1);
    }
    for (int kb = 0; kb < ROPE_DIM; kb += 4) {
        const int k = kb + koff;
        v2f a;  a.x = kr_row[k];       a.y = kr_row[k + 1];
        v2f b0; b0.x = q0[C_KV + k];   b0.y = q0[C_KV + k + 1];
        v2f b1; b1.x = q1[C_KV + k];   b1.y = q1[C_KV + k + 1];
        c0 = wmma_f32(a, b0, c0);
        c1 = wmma_f32(a, b1, c1);
    }
    const int mo = (lane >= 16) ? 8 : 0;
    #pragma unroll
    for (int j = 0; j < 8; ++j) {
        const int pos = mtile * 16 + j + mo;
        if (pos <= CACHE_LEN) {
            sp[((size_t)b * NUM_HEADS + lhalf) * S_PAD + pos] = c0[j];
            sp[((size_t)b * NUM_HEADS + 16 + lhalf) * S_PAD + pos] = c1[j];
        }
    }
}

// ---------------------------------------------------------------------------
// Kernel 4: softmax in place (+mask), zero the pad so PV can run full tiles.
// Grid 128 (b*32+h) x 256.
// ---------------------------------------------------------------------------
__global__ void k_softmax(const float* __restrict__ mask, float* __restrict__ sp) {
    const int bh = blockIdx.x, b = bh >> 5;
    float* row = sp + (size_t)bh * S_PAD;
    const float* mrow = mask + (size_t)b * S_NEW;
    __shared__ float red[256];
    float lmax = -INFINITY;
    for (int k = threadIdx.x; k < S_NEW; k += 256) {
        const float v = row[k] + mrow[k] * (-1e9f);
        row[k] = v;
        lmax = fmaxf(lmax, v);
    }
    red[threadIdx.x] = lmax; __syncthreads();
    for (int s = 128; s > 0; s >>= 1) {
        if (threadIdx.x < s) red[threadIdx.x] = fmaxf(red[threadIdx.x], red[threadIdx.x + s]);
        __syncthreads();
    }
    const float m = red[0];
    __syncthreads();
    float lsum = 0.f;
    for (int k = threadIdx.x; k < S_NEW; k += 256) {
        const float e = expf(row[k] - m);
        row[k] = e;
        lsum += e;
    }
    red[threadIdx.x] = lsum; __syncthreads();
    for (int s = 128; s > 0; s >>= 1) {
        if (threadIdx.x < s) red[threadIdx.x] += red[threadIdx.x + s];
        __syncthreads();
    }
    const float inv = 1.0f / red[0];
    __syncthreads();
    for (int k = threadIdx.x; k < S_PAD; k += 256)
        row[k] = (k < S_NEW) ? row[k] * inv : 0.0f;
}

// ---------------------------------------------------------------------------
// Kernel 5: o_lat = probs @ c_KV via WMMA. Per wave: one 16-feature tile x all
// 32 heads (2 accumulators share the B fragment). K = padded positions.
// Grid (4, B) x 256 (8 waves -> 32 ntiles).
// ---------------------------------------------------------------------------
__global__ void k_pv(const float* __restrict__ ckv,   // [B][4097][512]
                     const float* __restrict__ probs, // [B][32][4112], pad zeroed
                     float* __restrict__ olat) {      // [B][32][512]
    const int b = blockIdx.y;
    const int wave = threadIdx.x >> 5, lane = threadIdx.x & 31;
    const int ntile = blockIdx.x * 8 + wave;          // 0..31
    const int lhalf = lane & 15;
    const int koff = (lane >> 4) << 1;
    const float* p0 = probs + ((size_t)b * NUM_HEADS + lhalf) * S_PAD;
    const float* p1 = probs + ((size_t)b * NUM_HEADS + 16 + lhalf) * S_PAD;
    const int n = ntile * 16 + lhalf;
    const float* ckv_b = ckv + (size_t)b * S_NEW * C_KV;

    v8f c0 = {}; v8f c1 = {};
    for (int kb = 0; kb < S_PAD; kb += 4) {
        const int k = kb + koff;
        int r0 = k,     r1 = k + 1;
        if (r0 > CACHE_LEN) r0 = CACHE_LEN;           // probs there are zero
        if (r1 > CACHE_LEN) r1 = CACHE_LEN;
        v2f bf; bf.x = ckv_b[(size_t)r0 * C_KV + n];
                bf.y = ckv_b[(size_t)r1 * C_KV + n];
        v2f a0; a0.x = p0[k]; a0.y = p0[k + 1];
        v2f a1; a1.x = p1[k]; a1.y = p1[k + 1];
        c0 = wmma_f32(a0, bf, c0);
        c1 = wmma_f32(a1, bf, c1);
    }
    const int mo = (lane >= 16) ? 8 : 0;
    #pragma unroll
    for (int j = 0; j < 8; ++j) {
        const int h0 = j + mo;
        olat[((size_t)b * NUM_HEADS + h0) * C_KV + n] = c0[j];
        olat[((size_t)b * NUM_HEADS + 16 + h0) * C_KV + n] = c1[j];
    }
}

// ---------------------------------------------------------------------------
// Kernel 6: attn = o_lat @ W_UV (per head). Grid 32 (head) x 256.
// ---------------------------------------------------------------------------
__global__ void k_attn(const float* __restrict__ olat,
                       const float* __restrict__ W_UV,
                       float* __restrict__ attn) {    // [B][4096]
    const int h = blockIdx.x;
    __shared__ float ol[BATCH * C_KV];                // 8KB
    for (int i = threadIdx.x; i < BATCH * C_KV; i += 256)
        ol[i] = olat[((size_t)(i >> 9) * NUM_HEADS + h) * C_KV + (i & 511)];
    __syncthreads();
    const int d = threadIdx.x & 127, b0 = threadIdx.x >> 7;
    const float* wcol = W_UV + (size_t)h * HEAD_DIM + d;
    float a0 = 0.f, a1 = 0.f;
    for (int c = 0; c < C_KV; ++c) {
        const float w = wcol[(size_t)c * (NUM_HEADS * HEAD_DIM)];
        a0 += ol[b0 * C_KV + c] * w;
        a1 += ol[(b0 + 2) * C_KV + c] * w;
    }
    attn[(size_t)b0 * HIDDEN + h * HEAD_DIM + d] = a0;
    attn[(size_t)(b0 + 2) * HIDDEN + h * HEAD_DIM + d] = a1;
}

// ---------------------------------------------------------------------------
// Kernel 7: out = attn @ W_O. W_O (67MB) read exactly once, 4 batches reuse.
// Grid 16 x 256.
// ---------------------------------------------------------------------------
__global__ void k_out(const float* __restrict__ attn,
                      const float* __restrict__ W_O,
                      float* __restrict__ out) {
    __shared__ float at[BATCH * HIDDEN];              // 64KB
    for (int i = threadIdx.x; i < BATCH * HIDDEN; i += 256) at[i] = attn[i];
    __syncthreads();
    const int col = blockIdx.x * 256 + threadIdx.x;
    float acc[4] = {0.f, 0.f, 0.f, 0.f};
    for (int r = 0; r < HIDDEN; ++r) {
        const float w = W_O[(size_t)r * HIDDEN + col];
        #pragma unroll
        for (int b = 0; b < 4; ++b) acc[b] += at[b * HIDDEN + r] * w;
    }
    #pragma unroll
    for (int b = 0; b < 4; ++b) out[(size_t)b * HIDDEN + col] = acc[b];
}

// ---------------------------------------------------------------------------
extern "C" void kernel_launch(void* const* d_in, const int* in_sizes, int n_in,
                              void* d_out, int out_size, void* d_ws, size_t ws_size,
                              hipStream_t stream) {
    const float* hs    = (const float*)d_in[0];
    const float* mask  = (const float*)d_in[1];
    const float* ckv_c = (const float*)d_in[2];
    const float* kr_c  = (const float*)d_in[3];
    const float* W_DKV = (const float*)d_in[4];
    const float* W_UK  = (const float*)d_in[5];
    const float* W_UV  = (const float*)d_in[6];
    const float* W_DQ  = (const float*)d_in[7];
    const float* W_UQ  = (const float*)d_in[8];
    const float* W_QR  = (const float*)d_in[9];
    const float* W_KR  = (const float*)d_in[10];
    const float* W_O   = (const float*)d_in[11];

    float* out     = (float*)d_out;                      // [B][4096]
    float* out_ckv = out + (size_t)BATCH * HIDDEN;       // [B][4097][512]
    float* out_kr  = out_ckv + (size_t)BATCH * S_NEW * C_KV;  // [B][4097][64]

    float* ws    = (float*)d_ws;
    float* cq    = ws;                                   //  6144
    float* qcomb = cq + (size_t)BATCH * C_Q;             // 73728
    float* sp    = qcomb + (size_t)BATCH * NUM_HEADS * Q_DIM;  // 526336
    float* olat  = sp + (size_t)BATCH * NUM_HEADS * S_PAD;     // 65536
    float* attn  = olat + (size_t)BATCH * NUM_HEADS * C_KV;    // 16384

    k_down<<<9, 256, 0, stream>>>(hs, W_DKV, W_DQ, W_KR, cq, out_ckv, out_kr);
    k_copy_ckv<<<(BATCH * CACHE_LEN * 128) / 256, 256, 0, stream>>>(
        (const float4*)ckv_c, (float4*)out_ckv);
    k_copy_kr<<<(BATCH * CACHE_LEN * 16) / 256, 256, 0, stream>>>(
        (const float4*)kr_c, (float4*)out_kr);
    k_q<<<NUM_HEADS, 256, 0, stream>>>(cq, W_UQ, W_QR, W_UK, qcomb);
    k_scores<<<dim3(33, BATCH), 256, 0, stream>>>(out_ckv, out_kr, qcomb, sp);
    k_softmax<<<BATCH * NUM_HEADS, 256, 0, stream>>>(mask, sp);
    k_pv<<<dim3(4, BATCH), 256, 0, stream>>>(out_ckv, sp, olat);
    k_attn<<<NUM_HEADS, 256, 0, stream>>>(olat, W_UV, attn);
    k_out<<<HIDDEN / 256, 256, 0, stream>>>(attn, W_O, out);
}